// EpsModel_77799037600372
// MI455X (gfx1250) — compile-verified
//
#include <hip/hip_runtime.h>

// ---------------------------------------------------------------------------
// Batched 2x2 matrix polynomial:  ft = A0inv (I - t2)(I + t4)(I + t8)(I + t16)
// with t2 = x @ A0inv, t4 = t2^2, t8 = t4^2, t16 = t8^2.
//
// Bandwidth/VALU-balanced streaming kernel for MI455X (gfx1250):
//  - one float4 (b128) load + one float4 store per matrix, 4 matrices/thread
//  - all 2x2 math on float2 rows -> v_pk_mul_f32 / v_pk_fma_f32 / v_pk_add_f32
//    (exactly 40 packed VALU ops per matrix)
//  - A0inv broadcast to LDS once per block via the gfx1250 async engine
//    (global_load_async_to_lds_b128 + s_wait_asynccnt)
//  - unguarded fast kernel when nmat divides the tile (B = 4M does), guarded
//    fallback otherwise.
// ---------------------------------------------------------------------------

typedef float v2f __attribute__((ext_vector_type(2)));
typedef float v4f __attribute__((ext_vector_type(4)));
typedef int   v4i __attribute__((vector_size(16)));   // matches builtin param type

struct M2 { v2f r0, r1; };   // row-major 2x2: r0 = (m00, m01), r1 = (m10, m11)

// Z = X * Y : 2 pk_mul + 2 pk_fma after contraction
__device__ __forceinline__ M2 mmul(M2 X, M2 Y) {
    M2 Z;
    Z.r0 = X.r0.x * Y.r0 + X.r0.y * Y.r1;
    Z.r1 = X.r1.x * Y.r0 + X.r1.y * Y.r1;
    return Z;
}

#define TPB 256
#define EPT 4   // matrices per thread

// Broadcast A0inv (16 B) into LDS with the CDNA5 async-copy path.
__device__ __forceinline__ void load_A_to_lds(const float* A0inv, float* sA) {
#if __has_builtin(__builtin_amdgcn_global_load_async_to_lds_b128)
    if (threadIdx.x == 0) {
        auto* g = (__attribute__((address_space(1))) v4i*)A0inv;
        auto* l = (__attribute__((address_space(3))) v4i*)sA;
        __builtin_amdgcn_global_load_async_to_lds_b128(g, l, 0, 0);
    }
  #if __has_builtin(__builtin_amdgcn_s_wait_asynccnt)
    __builtin_amdgcn_s_wait_asynccnt(0);
  #else
    asm volatile("s_wait_asynccnt 0" ::: "memory");
  #endif
#else
    if (threadIdx.x < 4) sA[threadIdx.x] = A0inv[threadIdx.x];
#endif
    __syncthreads();
}

// Per-matrix math: 40 packed-f32 VALU ops.
__device__ __forceinline__ v4f eval_poly(v4f m, M2 A) {
    const v2f e0 = {1.0f, 0.0f};
    const v2f e1 = {0.0f, 1.0f};

    M2 X;
    X.r0.x = m.x; X.r0.y = m.y;
    X.r1.x = m.z; X.r1.y = m.w;

    M2 t2  = mmul(X,  A);      // 4
    M2 t4  = mmul(t2, t2);     // 4
    M2 t8  = mmul(t4, t4);     // 4
    M2 t16 = mmul(t8, t8);     // 4

    M2 u;                      // u = I - t2 : 2 pk ops
    u.r0 = e0 - t2.r0;
    u.r1 = e1 - t2.r1;

    M2 g = mmul(A, u);         // 4

    M2 w;
    w.r0 = t4.r0 + e0;  w.r1 = t4.r1 + e1;   g = mmul(g, w);  // 2 + 4
    w.r0 = t8.r0 + e0;  w.r1 = t8.r1 + e1;   g = mmul(g, w);  // 2 + 4
    w.r0 = t16.r0 + e0; w.r1 = t16.r1 + e1;  g = mmul(g, w);  // 2 + 4

    v4f r;
    r.x = g.r0.x; r.y = g.r0.y; r.z = g.r1.x; r.w = g.r1.y;
    return r;
}

// Fast path: grid exactly tiles nmat (nmat % (TPB*EPT) == 0). No bounds checks.
__global__ __launch_bounds__(TPB) void eps2x2_kernel_exact(const float* __restrict__ x,
                                                           const float* __restrict__ A0inv,
                                                           float* __restrict__ out) {
    __shared__ __align__(16) float sA[4];
    load_A_to_lds(A0inv, sA);

    M2 A;
    A.r0.x = sA[0]; A.r0.y = sA[1];
    A.r1.x = sA[2]; A.r1.y = sA[3];

    const v4f* __restrict__ px = (const v4f*)x;
    v4f* __restrict__ po = (v4f*)out;

    const int base = blockIdx.x * (TPB * EPT) + threadIdx.x;

    v4f xi[EPT];
#pragma unroll
    for (int e = 0; e < EPT; ++e) xi[e] = px[base + e * TPB];   // 4 b128 in flight

#pragma unroll
    for (int e = 0; e < EPT; ++e) po[base + e * TPB] = eval_poly(xi[e], A);
}

// Guarded fallback for non-divisible sizes.
__global__ __launch_bounds__(TPB) void eps2x2_kernel_guard(const float* __restrict__ x,
                                                           const float* __restrict__ A0inv,
                                                           float* __restrict__ out,
                                                           int nmat) {
    __shared__ __align__(16) float sA[4];
    load_A_to_lds(A0inv, sA);

    M2 A;
    A.r0.x = sA[0]; A.r0.y = sA[1];
    A.r1.x = sA[2]; A.r1.y = sA[3];

    const v4f* __restrict__ px = (const v4f*)x;
    v4f* __restrict__ po = (v4f*)out;

    const int base = blockIdx.x * (TPB * EPT) + threadIdx.x;
#pragma unroll
    for (int e = 0; e < EPT; ++e) {
        const int i = base + e * TPB;
        if (i < nmat) po[i] = eval_poly(px[i], A);
    }
}

extern "C" void kernel_launch(void* const* d_in, const int* in_sizes, int n_in,
                              void* d_out, int out_size, void* d_ws, size_t ws_size,
                              hipStream_t stream) {
    const float* x     = (const float*)d_in[0];
    const float* A0inv = (const float*)d_in[1];
    float* out         = (float*)d_out;

    const int nmat = in_sizes[0] / 4;               // [B,2,2] -> B matrices
    const int per_block = TPB * EPT;

    if (nmat % per_block == 0) {
        hipLaunchKernelGGL(eps2x2_kernel_exact, dim3(nmat / per_block), dim3(TPB),
                           0, stream, x, A0inv, out);
    } else {
        hipLaunchKernelGGL(eps2x2_kernel_guard,
                           dim3((nmat + per_block - 1) / per_block), dim3(TPB),
                           0, stream, x, A0inv, out, nmat);
    }
}